// ANIRepresentation_25769804390
// MI455X (gfx1250) — compile-verified
//
#include <hip/hip_runtime.h>
#include <hip/hip_bf16.h>
#include <math.h>

// ---------------- ANI constants (baked from reference) ----------------
#define S_NUM   7
#define NRBF    16
#define NPAIRS  28          // S*(S+1)/2
#define NA_     8
#define NZ_     4
#define SUB_    32          // NA*NZ
#define ROW_    1008        // S*NRBF + NPAIRS*SUB = 112 + 896
#define ANG_OFF 112

__device__ __forceinline__ float kRC()    { return 0.51f; }
__device__ __forceinline__ float kRMIN()  { return 0.08f; }
__device__ __forceinline__ float kRCA()   { return 0.35f; }
__device__ __forceinline__ float kRAMIN() { return 0.08f; }
__device__ __forceinline__ float kETAR()  { return 1970.0f; }
__device__ __forceinline__ float kETAA()  { return 1250.0f; }
__device__ __forceinline__ float kZETA()  { return 14.1f; }
__device__ __forceinline__ float kPI()    { return 3.14159265358979323846f; }

// ---------------- CDNA5 async-to-LDS helpers (probe-guarded) ----------------
// Diagnostic from round 1 shows the builtin's params are addrspace-qualified
// int* ("__device__ int *" == addrspace(1) int*). Cast accordingly.
typedef __attribute__((address_space(1))) int  g_int;
typedef __attribute__((address_space(3))) int  l_int;

__device__ __forceinline__ void async_copy_b32(const void* gsrc, void* lds_dst) {
#if __has_builtin(__builtin_amdgcn_global_load_async_to_lds_b32)
  __builtin_amdgcn_global_load_async_to_lds_b32(
      (g_int*)gsrc, (l_int*)lds_dst, /*offset=*/0, /*cpol=*/0);
#else
  *(int*)lds_dst = *(const int*)gsrc;   // synchronous fallback
#endif
}

__device__ __forceinline__ void wait_async0() {
#if __has_builtin(__builtin_amdgcn_s_wait_asynccnt)
  __builtin_amdgcn_s_wait_asynccnt(0);
#else
  asm volatile("s_wait_asynccnt 0" ::: "memory");
#endif
}

__device__ __forceinline__ float cutoff_f(float d, float rc) {
  return (d < rc) ? (0.5f * cosf(kPI() * d / rc) + 0.5f) : 0.0f;
}

// ---------------- zero-init of d_out (harness poisons it) ----------------
__global__ __launch_bounds__(256) void ani_zero_kernel(float* __restrict__ out, int n) {
  int idx = blockIdx.x * blockDim.x + threadIdx.x;
  int n4 = n >> 2;
  if (idx < n4) {
    ((float4*)out)[idx] = make_float4(0.f, 0.f, 0.f, 0.f);
  }
  int tail = n & 3;
  if (idx < tail) out[(n4 << 2) + idx] = 0.f;
}

// ---------------- radial pass: 16 lanes per pair, LDS-staged stream -------
// pair stream (i, j, d) is staged into LDS with async-to-LDS loads, then each
// half-wave (16 lanes) computes the 16 RBF features of one pair and issues a
// contiguous 64B wave-coalesced atomic add into two output rows.
#define RAD_TILE 256
__global__ __launch_bounds__(256) void ani_radial_kernel(
    const int*   __restrict__ atom_index,
    const int*   __restrict__ pair_ind,   // [2*P]: i at [p], j at [P+p]
    const float* __restrict__ d_ij,       // [P]
    float*       __restrict__ out,
    int P)
{
  __shared__ int   s_i[RAD_TILE];
  __shared__ int   s_j[RAD_TILE];
  __shared__ float s_d[RAD_TILE];

  const int tid  = threadIdx.x;
  const int base = blockIdx.x * RAD_TILE;
  const int p    = base + tid;

  if (p < P) {                       // EXEC-masked per-lane async transfers
    async_copy_b32(pair_ind + p,     &s_i[tid]);
    async_copy_b32(pair_ind + P + p, &s_j[tid]);
    async_copy_b32(d_ij + p,         &s_d[tid]);
  }
  wait_async0();
  __syncthreads();

  const int half = tid >> 4;         // which pair of the 16 handled this pass
  const int k    = tid & 15;         // RBF index
  const float step   = (kRC() - kRMIN()) / (float)NRBF;
  const float center = kRMIN() + (float)k * step;

#pragma unroll 4
  for (int q = 0; q < RAD_TILE / 16; ++q) {
    const int lp = q * 16 + half;
    const int pp = base + lp;
    if (pp < P) {
      const int   i  = s_i[lp];
      const int   j  = s_j[lp];
      const float d  = s_d[lp];
      const int   si = atom_index[i];   // wave-uniform per half: merged load
      const int   sj = atom_index[j];

      const float dd   = d - center;
      const float term = 0.25f * expf(-kETAR() * dd * dd) * cutoff_f(d, kRC());

      atomicAdd(&out[(size_t)i * ROW_ + (size_t)(sj * NRBF + k)], term);
      atomicAdd(&out[(size_t)j * ROW_ + (size_t)(si * NRBF + k)], term);
    }
  }
}

// ---------------- angular pass: one wave32 per triple ---------------------
// lane = a*4 + z indexes the 8x4 outer product f2(a) (x) f1(z); the 32-float
// row update becomes a single contiguous 128B wave-coalesced atomic add.
__global__ __launch_bounds__(256) void ani_angular_kernel(
    const int*   __restrict__ atom_index,
    const int*   __restrict__ pair_ind,   // [2*P]
    const float* __restrict__ r_ij,       // [P*3]
    const int*   __restrict__ central,    // [T]
    const int*   __restrict__ pidx12,     // [2*T]
    const int*   __restrict__ sign12,     // [2*T]
    float*       __restrict__ out,
    int T, int P)
{
  const int wave = (blockIdx.x * blockDim.x + threadIdx.x) >> 5;
  const int lane = threadIdx.x & 31;
  if (wave >= T) return;
  const int t = wave;

  // ---- wave-uniform scalars (same address across lanes -> merged loads) ----
  const int c   = central[t];
  const int p0  = pidx12[t];
  const int p1  = pidx12[T + t];
  const int sg0 = sign12[t];
  const int sg1 = sign12[T + t];
  const float f0 = (float)sg0;
  const float f1 = (float)sg1;

  const float v0x = r_ij[3 * p0 + 0] * f0;
  const float v0y = r_ij[3 * p0 + 1] * f0;
  const float v0z = r_ij[3 * p0 + 2] * f0;
  const float v1x = r_ij[3 * p1 + 0] * f1;
  const float v1y = r_ij[3 * p1 + 1] * f1;
  const float v1z = r_ij[3 * p1 + 2] * f1;

  const float d0 = sqrtf(v0x * v0x + v0y * v0y + v0z * v0z);
  const float d1 = sqrtf(v1x * v1x + v1y * v1y + v1z * v1z);

  const float cos_a = 0.95f * (v0x * v1x + v0y * v1y + v0z * v1z) / (d0 * d1);
  const float ang   = acosf(cos_a);

  const float fcj  = cutoff_f(d0, kRCA()) * cutoff_f(d1, kRCA());
  const float davg = 0.5f * (d0 + d1);

  // species of the non-central atoms of each pair
  const int i0 = pair_ind[p0], j0 = pair_ind[P + p0];
  const int i1 = pair_ind[p1], j1 = pair_ind[P + p1];
  const int s0 = (sg0 == 1) ? atom_index[j0] : atom_index[i0];
  const int s1 = (sg1 == 1) ? atom_index[j1] : atom_index[i1];

  const int a = (s0 < s1) ? s0 : s1;
  const int b = (s0 < s1) ? s1 : s0;
  const int triu = a * S_NUM - (a * (a - 1)) / 2 + (b - a);

  // ---- per-lane outer-product term ----
  const int ia = lane >> 2;   // shfa index 0..7
  const int iz = lane & 3;    // shfz index 0..3
  const float shfa = kRAMIN() + (float)ia * ((kRCA() - kRAMIN()) / (float)NA_);
  const float shfz = ((float)iz + 0.5f) * (kPI() / (float)NZ_);

  const float x   = (1.0f + cosf(ang - shfz)) * 0.5f;
  const float fz  = powf(x, kZETA());
  const float dm  = davg - shfa;
  const float fr  = expf(-kETAA() * dm * dm);
  const float term = 2.0f * fr * fz * fcj;

  atomicAdd(&out[(size_t)c * ROW_ + (size_t)(ANG_OFF + triu * SUB_ + lane)], term);
}

// ---------------- launcher ----------------
extern "C" void kernel_launch(void* const* d_in, const int* in_sizes, int n_in,
                              void* d_out, int out_size, void* d_ws, size_t ws_size,
                              hipStream_t stream) {
  const int*   atom_index = (const int*)d_in[0];    // [N]
  const int*   pair_ind   = (const int*)d_in[1];    // [2*P]
  const float* d_ij       = (const float*)d_in[2];  // [P]
  const float* r_ij       = (const float*)d_in[3];  // [P*3]
  const int*   central    = (const int*)d_in[4];    // [T]
  const int*   pidx12     = (const int*)d_in[5];    // [2*T]
  const int*   sign12     = (const int*)d_in[6];    // [2*T]
  float* out = (float*)d_out;

  const int P = in_sizes[2];
  const int T = in_sizes[4];

  // 1) zero the accumulator/output
  {
    int n4 = out_size >> 2;
    int blocks = (n4 + 255) / 256;
    if (blocks < 1) blocks = 1;
    ani_zero_kernel<<<blocks, 256, 0, stream>>>(out, out_size);
  }
  // 2) radial scatter-accumulate
  {
    int blocks = (P + RAD_TILE - 1) / RAD_TILE;
    ani_radial_kernel<<<blocks, 256, 0, stream>>>(atom_index, pair_ind, d_ij, out, P);
  }
  // 3) angular scatter-accumulate (8 waves = 8 triples per block)
  {
    int blocks = (T + 7) / 8;
    ani_angular_kernel<<<blocks, 256, 0, stream>>>(atom_index, pair_ind, r_ij,
                                                   central, pidx12, sign12, out, T, P);
  }
}